// SpearmanLoss_81252191306070
// MI455X (gfx1250) — compile-verified
//
#include <hip/hip_runtime.h>
#include <hip/hip_bf16.h>

// ---------------------------------------------------------------------------
// SpearmanLoss on MI455X (gfx1250, wave32)
//   pred, target: [N=65536, M=32] f32. Rank each column (soft_rank, l2, s=1),
//   then global-center/normalize both rank fields and output -(p.t)/100.
//
// Pipeline (all on `stream`, workspace-resident, L2-sized working set):
//   1) pack_keys   : u64 key = (~asc_sortable(f32) << 32) | index  (desc order,
//                    stable ties)  -> 64 jobs x 65536 keys = 32 MB (fits L2)
//   2) bitonic_sort: 1 workgroup (1024 thr = 32 waves) per job, 136 passes
//   3) pava_expand : y = s - [n..1] staged in 256 KB LDS (CDNA5 320KB/WGP),
//                    thread-0 sequential PAVA, parallel block expand+scatter
//   4) reduce_wmma : TDM (tensor_load_to_lds) stages per-wave 2KB slices of
//                    each rank field into LDS; s_wait_tensorcnt; then five
//                    global sums via V_WMMA_F32_16X16X4_F32 fed from LDS
//                    (f32 MACs; ranks span [1,65536] -> low precision is out)
//   5) finalize    : scalar loss
// ---------------------------------------------------------------------------

#define NN 65536
#define MM 32
#define JOBS 64              // 32 pred columns + 32 target columns
#define MN (MM * NN)         // 2097152 elements per rank field
#define TILE 4096            // floats per workgroup per stream per iteration
#define WSLICE 512           // floats per wave per stream (2 KB)
#define NTILES (MN / TILE)   // 512

typedef __attribute__((ext_vector_type(2))) float v2f;
typedef __attribute__((ext_vector_type(8))) float v8f;
typedef __attribute__((ext_vector_type(4))) unsigned int v4u;
typedef __attribute__((ext_vector_type(8))) unsigned int v8u;

// --- float <-> descending-sortable key helpers ------------------------------
__device__ __forceinline__ unsigned int f32_to_desc(float v) {
    unsigned int u = __float_as_uint(v);
    unsigned int asc = (u & 0x80000000u) ? ~u : (u | 0x80000000u);
    return ~asc;                       // ascending u32 == descending float
}
__device__ __forceinline__ float key_to_f32(unsigned long long k) {
    unsigned int asc = ~(unsigned int)(k >> 32);
    unsigned int orig = (asc & 0x80000000u) ? (asc ^ 0x80000000u) : ~asc;
    return __uint_as_float(orig);
}

// --- TDM: 1-D tile DMA, global -> LDS (D# per CDNA5 ISA) --------------
__device__ __forceinline__ void tdm_load_1d(unsigned lds_addr,
                                            unsigned long long gaddr,
                                            unsigned nelem) {
    v4u g0;
    g0.x = 1u;                                        // count=1 (valid), user D#
    g0.y = lds_addr;                                  // lds_addr [63:32]
    g0.z = (unsigned)(gaddr & 0xFFFFFFFFull);         // global_addr [95:64]
    g0.w = (unsigned)((gaddr >> 32) & 0x01FFFFFFull)  // global_addr [120:96]
         | (2u << 30);                                // type=2 ("image")
    v8u g1;
    const unsigned tdim0 = 1u << 22;                  // generous 1-D extent
    g1[0] = (2u << 16);                               // data_size=4B, no mcast
    g1[1] = (tdim0 & 0xFFFFu) << 16;                  // tensor_dim0[15:0]
    g1[2] = ((tdim0 >> 16) & 0xFFFFu) | (1u << 16);   // tensor_dim0[31:16], tensor_dim1=1
    g1[3] = (nelem & 0xFFFFu) << 16;                  // tile_dim0
    g1[4] = 0u;                                       // tile_dim1/2 unused
    g1[5] = tdim0;                                    // tensor_dim0_stride[31:0]
    g1[6] = 0u;
    g1[7] = 0u;
    asm volatile("tensor_load_to_lds %0, %1" :: "s"(g0), "s"(g1) : "memory");
}

// --- 1) build sort keys -----------------------------------------------------
__global__ __launch_bounds__(256) void pack_keys(
        const float* __restrict__ pred, const float* __restrict__ tgt,
        unsigned long long* __restrict__ K) {
    unsigned int gid = blockIdx.x * blockDim.x + threadIdx.x; // [0, JOBS*NN)
    unsigned int job = gid >> 16;          // /NN
    unsigned int i   = gid & (NN - 1);     // %NN
    unsigned int m   = job & 31;
    const float* src = (job < 32) ? pred : tgt;
    float v = src[(size_t)i * MM + m];     // column m, row i of [N,M]
    K[gid] = ((unsigned long long)f32_to_desc(v) << 32) | i;
}

// --- 2) in-L2 bitonic sort, one workgroup per job ---------------------------
__global__ __launch_bounds__(1024) void bitonic_sort(unsigned long long* __restrict__ Kall) {
    unsigned long long* K = Kall + (size_t)blockIdx.x * NN;
    const int tid = threadIdx.x;
    for (int k = 2; k <= NN; k <<= 1) {
        for (int j = k >> 1; j > 0; j >>= 1) {
            for (int t = tid; t < NN / 2; t += 1024) {
                int i   = ((t & ~(j - 1)) << 1) | (t & (j - 1));
                int ixj = i | j;
                bool up = ((i & k) == 0);  // ascending u64 == descending float
                unsigned long long a = K[i], b = K[ixj];
                if ((a > b) == up) { K[i] = b; K[ixj] = a; }
            }
            __threadfence();
            __syncthreads();
        }
    }
}

// --- 3) PAVA (non-increasing) + expand + scatter ----------------------------
__global__ __launch_bounds__(256) void pava_expand(
        const unsigned long long* __restrict__ Kall,
        float* __restrict__ Rall,
        float* __restrict__ sumsAll,
        float* __restrict__ cntsAll) {
    extern __shared__ float y[];           // NN floats = 256 KB LDS (CDNA5 WGP)
    const int job = blockIdx.x;
    const unsigned long long* K = Kall + (size_t)job * NN;
    float* R    = Rall    + (size_t)job * NN;
    float* sums = sumsAll + (size_t)job * NN;
    float* cnts = cntsAll + (size_t)job * NN;

    // stage y[i] = s[i] - (N - i) in LDS  (w = [n, n-1, ..., 1])
    for (int i = threadIdx.x; i < NN; i += blockDim.x) {
        __builtin_prefetch(&K[i + 4 * blockDim.x], 0, 1);
        y[i] = key_to_f32(K[i]) - (float)(NN - i);
    }
    __syncthreads();

    __shared__ int s_top;
    if (threadIdx.x == 0) {
        int top = 0;
        for (int i = 0; i < NN; ++i) {
            sums[top] = y[i];
            cnts[top] = 1.0f;
            ++top;
            while (top >= 2 &&
                   sums[top - 1] * cnts[top - 2] >= sums[top - 2] * cnts[top - 1]) {
                sums[top - 2] += sums[top - 1];
                cnts[top - 2] += cnts[top - 1];
                --top;
            }
        }
        // in-place: sums -> block means, cnts -> block start indices (int)
        int* starts = (int*)cnts;
        int run = 0;
        for (int b = 0; b < top; ++b) {
            float c = cnts[b];
            sums[b] = sums[b] / c;
            starts[b] = run;
            run += (int)c;
        }
        s_top = top;
        __threadfence();
    }
    __syncthreads();

    const int top = s_top;
    const int* starts = (const int*)cnts;
    for (int i = threadIdx.x; i < NN; i += blockDim.x) {
        int lo = 0, hi = top - 1;                 // largest b: starts[b] <= i
        while (lo < hi) {
            int mid = (lo + hi + 1) >> 1;
            if (starts[mid] <= i) lo = mid; else hi = mid - 1;
        }
        unsigned long long k = K[i];
        float rank = key_to_f32(k) - sums[lo];    // ranks_sorted = s - v
        unsigned int sigma = (unsigned int)(k & 0xFFFFFFFFu);
        R[sigma] = rank;                          // scatter to original order
    }
}

// --- 4) five global sums: TDM -> LDS -> V_WMMA_F32_16X16X4_F32 --------------
__global__ void init_acc(float* __restrict__ acc) {
    if (threadIdx.x < 8) acc[threadIdx.x] = 0.0f;
}

__global__ __launch_bounds__(256) void reduce_wmma(
        const float* __restrict__ R, float* __restrict__ acc) {
    __shared__ float shP[TILE];
    __shared__ float shT[TILE];
    const float* __restrict__ P = R;
    const float* __restrict__ T = R + (size_t)MN;
    const int lane = threadIdx.x & 31;
    const int wwg  = threadIdx.x >> 5;           // wave id in WG [0,8)

    // per-wave private LDS slices -> no workgroup barriers needed at all
    const unsigned ldsP = (unsigned)(uintptr_t)&shP[0] + (unsigned)(wwg * WSLICE * 4);
    const unsigned ldsT = (unsigned)(uintptr_t)&shT[0] + (unsigned)(wwg * WSLICE * 4);

    v8f cp = {}, ct = {}, cpp = {}, ctt = {}, cpt = {};
    v2f ones; ones.x = 1.0f; ones.y = 1.0f;      // B = 4x16 all-ones

    for (int tile = blockIdx.x; tile < NTILES; tile += gridDim.x) {
        const size_t off = (size_t)tile * TILE + (size_t)wwg * WSLICE;
        // TDM is unordered vs DS ops: drain our DS reads before LDS reuse
        asm volatile("s_wait_dscnt 0x0" ::: "memory");
        tdm_load_1d(ldsP, (unsigned long long)(uintptr_t)(P + off), WSLICE);
        tdm_load_1d(ldsT, (unsigned long long)(uintptr_t)(T + off), WSLICE);
        __builtin_amdgcn_s_wait_tensorcnt(0);    // both DMAs of this wave done

        for (int c = 0; c < WSLICE / 64; ++c) {  // 8 wmma chunks per slice
            int idx = wwg * WSLICE + c * 64 + lane * 2;
            v2f p, t, pp, tt, pt;
            p.x = shP[idx]; p.y = shP[idx + 1];
            t.x = shT[idx]; t.y = shT[idx + 1];
            pp.x = p.x * p.x; pp.y = p.y * p.y;
            tt.x = t.x * t.x; tt.y = t.y * t.y;
            pt.x = p.x * t.x; pt.y = p.y * t.y;
            cp  = __builtin_amdgcn_wmma_f32_16x16x4_f32(false, p,  false, ones, (short)0, cp,  false, false);
            ct  = __builtin_amdgcn_wmma_f32_16x16x4_f32(false, t,  false, ones, (short)0, ct,  false, false);
            cpp = __builtin_amdgcn_wmma_f32_16x16x4_f32(false, pp, false, ones, (short)0, cpp, false, false);
            ctt = __builtin_amdgcn_wmma_f32_16x16x4_f32(false, tt, false, ones, (short)0, ctt, false, false);
            cpt = __builtin_amdgcn_wmma_f32_16x16x4_f32(false, pt, false, ones, (short)0, cpt, false, false);
        }
    }

    // each D column n holds the full running total split across lane n / n+16
    v8f a5[5] = {cp, ct, cpp, ctt, cpt};
    float s5[5];
    for (int q = 0; q < 5; ++q) {
        v8f a = a5[q];
        float s = a[0] + a[1] + a[2] + a[3] + a[4] + a[5] + a[6] + a[7];
        s += __shfl_down(s, 16);          // wave32: lane L += lane L+16
        s5[q] = s;
    }
    if (lane == 0)
        for (int q = 0; q < 5; ++q) atomicAdd(&acc[q], s5[q]);
}

// --- 5) scalar finish -------------------------------------------------------
__global__ void finalize(const float* __restrict__ acc, float* __restrict__ out) {
    if (threadIdx.x == 0 && blockIdx.x == 0) {
        double n   = (double)MN;
        double Sp  = acc[0], St = acc[1], Spp = acc[2], Stt = acc[3], Spt = acc[4];
        double dot = Spt - Sp * St / n;
        double np_ = sqrt(Spp - Sp * Sp / n);
        double nt_ = sqrt(Stt - St * St / n);
        out[0] = (float)(-(dot / (np_ * nt_)) / 100.0);
    }
}

// ---------------------------------------------------------------------------
extern "C" void kernel_launch(void* const* d_in, const int* in_sizes, int n_in,
                              void* d_out, int out_size, void* d_ws, size_t ws_size,
                              hipStream_t stream) {
    (void)in_sizes; (void)n_in; (void)out_size; (void)ws_size;
    const float* pred = (const float*)d_in[0];
    const float* tgt  = (const float*)d_in[1];

    char* ws = (char*)d_ws;
    // workspace layout (bytes): keys 32MB | ranks 16MB | sums 16MB | cnts 16MB | acc
    unsigned long long* K = (unsigned long long*)(ws);
    float* R    = (float*)(ws + ((size_t)32 << 20));
    float* sums = (float*)(ws + ((size_t)48 << 20));
    float* cnts = (float*)(ws + ((size_t)64 << 20));
    float* acc  = (float*)(ws + ((size_t)80 << 20));

    pack_keys  <<<(JOBS * NN) / 256, 256, 0, stream>>>(pred, tgt, K);
    bitonic_sort<<<JOBS, 1024, 0, stream>>>(K);
    pava_expand<<<JOBS, 256, NN * sizeof(float), stream>>>(K, R, sums, cnts);
    init_acc   <<<1, 32, 0, stream>>>(acc);
    reduce_wmma<<<128, 256, 0, stream>>>(R, acc);
    finalize   <<<1, 1, 0, stream>>>(acc, (float*)d_out);
}